// Group_88252987998631
// MI455X (gfx1250) — compile-verified
//
#include <hip/hip_runtime.h>

typedef float v2f __attribute__((ext_vector_type(2)));
typedef float v8f __attribute__((ext_vector_type(8)));

#define NPTS 8192
#define NG   512
#define KNN  32
#define GT   16          // groups per block
#define PT   256         // points per tile
#define NT   (NPTS / PT) // 32 tiles

// ---------------------------------------------------------------------------
// Kernel A: farthest point sampling. One block per batch, 1024 threads
// (32 wave32s), each thread owns 8 points entirely in registers.
// Writes centers [B, 512, 3] directly into the d_out center region.
// ---------------------------------------------------------------------------
__global__ __launch_bounds__(1024) void fps_kernel(const float* __restrict__ xyz,
                                                   float* __restrict__ centers) {
    __shared__ float sc[3];
    __shared__ int   sFar;
    __shared__ float redV[32];
    __shared__ int   redI[32];

    const int b    = blockIdx.x;
    const int tid  = threadIdx.x;
    const int lane = tid & 31;
    const int w    = tid >> 5;

    float X[8], Y[8], Z[8], D[8];
    const float* base = xyz + (size_t)b * NPTS * 3 + (size_t)tid * 8 * 3;
#pragma unroll
    for (int s = 0; s < 8; ++s) {
        X[s] = base[s * 3 + 0];
        Y[s] = base[s * 3 + 1];
        Z[s] = base[s * 3 + 2];
        D[s] = 1e10f;
    }
    if (tid == 0) sFar = 0;
    __syncthreads();

    for (int i = 0; i < NG; ++i) {
        const int f     = sFar;
        const int owner = f >> 3;
        const int slot  = f & 7;
        if (tid == owner) {
            float cx = 0.f, cy = 0.f, cz = 0.f;
#pragma unroll
            for (int s = 0; s < 8; ++s)
                if (s == slot) { cx = X[s]; cy = Y[s]; cz = Z[s]; }
            sc[0] = cx; sc[1] = cy; sc[2] = cz;
            float* o = centers + ((size_t)b * NG + i) * 3;
            o[0] = cx; o[1] = cy; o[2] = cz;
        }
        __syncthreads();
        const float cx = sc[0], cy = sc[1], cz = sc[2];

        float bv = -1.0f;
        int   bi = 0;
#pragma unroll
        for (int s = 0; s < 8; ++s) {
            float dx = X[s] - cx, dy = Y[s] - cy, dz = Z[s] - cz;
            float d  = dx * dx + dy * dy + dz * dz;
            D[s] = fminf(D[s], d);
            if (D[s] > bv) { bv = D[s]; bi = tid * 8 + s; }   // first-max tiebreak
        }
        // wave32 argmax, tie -> lower index
#pragma unroll
        for (int off = 16; off > 0; off >>= 1) {
            float ov = __shfl_xor(bv, off, 32);
            int   oi = __shfl_xor(bi, off, 32);
            if (ov > bv || (ov == bv && oi < bi)) { bv = ov; bi = oi; }
        }
        if (lane == 0) { redV[w] = bv; redI[w] = bi; }
        __syncthreads();
        if (w == 0) {
            float v  = redV[lane];
            int   id = redI[lane];
#pragma unroll
            for (int off = 16; off > 0; off >>= 1) {
                float ov = __shfl_xor(v, off, 32);
                int   oi = __shfl_xor(id, off, 32);
                if (ov > v || (ov == v && oi < id)) { v = ov; id = oi; }
            }
            if (lane == 0) sFar = id;
        }
        __syncthreads();
    }
}

// ---------------------------------------------------------------------------
// Kernel B: fused distance (WMMA f32 16x16x4) + top-32 selection + gather.
// One block = 16 groups of one batch, 512 threads (16 waves).
// Wave w computes the 16x16 dot tile for points [w*16, w*16+16) each step,
// then owns group w for selection. Top-32 list lives one entry per lane.
// ---------------------------------------------------------------------------
__global__ __launch_bounds__(512) void knn_kernel(const float* __restrict__ xyz,
                                                  const float* __restrict__ color,
                                                  const float* __restrict__ centers,
                                                  float* __restrict__ neigh,
                                                  float* __restrict__ feat) {
    __shared__ float ctr[GT][4];      // (x,y,z,0)
    __shared__ float cc[GT];          // |c|^2
    __shared__ float pts[PT][4];      // (x,y,z,|p|^2)  -- K=3 slot of A is 0, so safe
    __shared__ float distbuf[GT][PT];

    const int tid  = threadIdx.x;
    const int lane = tid & 31;
    const int w    = tid >> 5;
    const int b    = blockIdx.x / (NG / GT);
    const int gt   = blockIdx.x % (NG / GT);
    const int g0   = gt * GT;

    if (tid < GT) {
        const float* c = centers + ((size_t)(b * NG + g0 + tid)) * 3;
        float x = c[0], y = c[1], z = c[2];
        ctr[tid][0] = x; ctr[tid][1] = y; ctr[tid][2] = z; ctr[tid][3] = 0.f;
        cc[tid] = x * x + y * y + z * z;
    }

    float topD = __builtin_inff();
    int   topI = 0x40000000 + lane;   // unique sentinels -> strict total order

    const int m  = lane & 15;
    const int hi = lane >> 4;
    const int kb = hi * 2;

    for (int t = 0; t < NT; ++t) {
        const int pbase = t * PT;
        __syncthreads();
        if (tid < PT) {
            const float* p = xyz + ((size_t)(b * NPTS + pbase + tid)) * 3;
            float x = p[0], y = p[1], z = p[2];
            pts[tid][0] = x; pts[tid][1] = y; pts[tid][2] = z;
            pts[tid][3] = x * x + y * y + z * z;
            if (t + 1 < NT) __builtin_prefetch(p + PT * 3, 0, 1);
        }
        __syncthreads();

        // --- WMMA 16x16 dot tile (EXEC is full here: uniform control flow) ---
        const int pn = w * 16 + m;
        v2f av, bv;
        av.x = ctr[m][kb];  av.y = ctr[m][kb + 1];
        bv.x = pts[pn][kb]; bv.y = pts[pn][kb + 1];
        v8f acc = {0.f, 0.f, 0.f, 0.f, 0.f, 0.f, 0.f, 0.f};
        acc = __builtin_amdgcn_wmma_f32_16x16x4_f32(
            /*neg_a=*/false, av, /*neg_b=*/false, bv,
            /*c_mod=*/(short)0, acc, /*reuse_a=*/false, /*reuse_b=*/false);
        const float pp = pts[pn][3];
#pragma unroll
        for (int v = 0; v < 8; ++v) {
            int M = v + 8 * hi;
            distbuf[M][w * 16 + m] = cc[M] + pp - 2.0f * acc[v];
        }
        __syncthreads();

        // --- selection: wave w merges its group's 256 candidates ---
        float* row = distbuf[w];
        for (int it = 0; it < KNN; ++it) {
            // tile argmin (tie -> lower slot == lower global index)
            float mn = __builtin_inff();
            int   ms = 1 << 30;
#pragma unroll
            for (int j = 0; j < PT / 32; ++j) {
                int   s = lane + j * 32;
                float d = row[s];
                if (d < mn) { mn = d; ms = s; }
            }
#pragma unroll
            for (int off = 16; off > 0; off >>= 1) {
                float od = __shfl_xor(mn, off, 32);
                int   os = __shfl_xor(ms, off, 32);
                if (od < mn || (od == mn && os < ms)) { mn = od; ms = os; }
            }
            const int cand = pbase + ms;
            // current worst of top-32 (one entry per lane)
            float mxD = topD; int mxI = topI; int mxL = lane;
#pragma unroll
            for (int off = 16; off > 0; off >>= 1) {
                float od = __shfl_xor(mxD, off, 32);
                int   oi = __shfl_xor(mxI, off, 32);
                int   ol = __shfl_xor(mxL, off, 32);
                if (od > mxD || (od == mxD && oi > mxI)) { mxD = od; mxI = oi; mxL = ol; }
            }
            const bool better = (mn < mxD) || (mn == mxD && cand < mxI);
            if (!better) break;                 // wave-uniform
            if (lane == mxL) { topD = mn; topI = cand; }
            if (lane == (ms & 31)) row[ms] = __builtin_inff();  // consume
        }
        __syncthreads();
    }

    // --- bitonic sort of (dist, idx) across the wave: ascending distance ---
#pragma unroll
    for (int kk = 2; kk <= 32; kk <<= 1) {
#pragma unroll
        for (int j = kk >> 1; j > 0; j >>= 1) {
            float oD = __shfl_xor(topD, j, 32);
            int   oI = __shfl_xor(topI, j, 32);
            bool up          = ((lane & kk) == 0);
            bool lower       = ((lane & j) == 0);
            bool mineSmaller = (topD < oD) || (topD == oD && topI < oI);
            bool keepMine    = (lower == up) ? mineSmaller : !mineSmaller;
            if (!keepMine) { topD = oD; topI = oI; }
        }
    }

    // --- gather + write outputs: wave w -> group g0+w, lane L -> slot L ---
    {
        const int    g   = g0 + w;
        const int    idx = topI;
        const float* p   = xyz   + ((size_t)b * NPTS + idx) * 3;
        const float* cl  = color + ((size_t)b * NPTS + idx) * 3;
        float nx = p[0] - ctr[w][0];
        float ny = p[1] - ctr[w][1];
        float nz = p[2] - ctr[w][2];
        float c0 = cl[0], c1 = cl[1], c2 = cl[2];
        size_t nb = (((size_t)b * NG + g) * KNN + lane) * 3;
        neigh[nb + 0] = nx; neigh[nb + 1] = ny; neigh[nb + 2] = nz;
        size_t fb = (((size_t)b * NG + g) * KNN + lane) * 6;
        feat[fb + 0] = nx; feat[fb + 1] = ny; feat[fb + 2] = nz;
        feat[fb + 3] = c0; feat[fb + 4] = c1; feat[fb + 5] = c2;
    }
}

// ---------------------------------------------------------------------------
extern "C" void kernel_launch(void* const* d_in, const int* in_sizes, int n_in,
                              void* d_out, int out_size, void* d_ws, size_t ws_size,
                              hipStream_t stream) {
    (void)in_sizes; (void)n_in; (void)d_ws; (void)ws_size; (void)out_size;
    const float* xyz   = (const float*)d_in[0];
    const float* color = (const float*)d_in[1];
    float* out = (float*)d_out;

    const int B = 32;
    // d_out layout (flat, return order): neighborhood, center, features
    float* neigh   = out;                                   // [B,512,32,3]
    float* centers = neigh + (size_t)B * NG * KNN * 3;      // [B,512,3]
    float* feat    = centers + (size_t)B * NG * 3;          // [B,512,32,6]

    fps_kernel<<<B, 1024, 0, stream>>>(xyz, centers);
    knn_kernel<<<B * (NG / GT), 512, 0, stream>>>(xyz, color, centers, neigh, feat);
}